// ReMultiHeadSelfAttentionV2_4758823764688
// MI455X (gfx1250) — compile-verified
//
#include <hip/hip_runtime.h>

// ---------------------------------------------------------------------------
// Problem constants (match reference)
// ---------------------------------------------------------------------------
constexpr int DIM = 128;
constexpr int H   = 8;
constexpr int DH  = 16;
constexpr int P   = 32;
constexpr int S   = 512;
constexpr int Bn  = 64;
constexpr int F   = P + S;        // 544
constexpr int TS  = S * Bn;       // 32768 sequence rows (s*B + b)
constexpr int T   = TS + P;       // 32800 virtual rows for QKV GEMM

// workspace layout (float offsets)
constexpr int QS_OFF  = 0;
constexpr int KS_OFF  = QS_OFF + Bn * H * S * DH;   //  4 Mi floats
constexpr int VS_OFF  = KS_OFF + Bn * H * S * DH;
constexpr int QP_OFF  = VS_OFF + Bn * H * S * DH;   // 12 Mi floats
constexpr int KP_OFF  = QP_OFF + H * P * DH;
constexpr int VP_OFF  = KP_OFF + H * P * DH;
constexpr int ATT_OFF = VP_OFF + H * P * DH;        // + F*Bn*DIM floats

typedef float v2f __attribute__((ext_vector_type(2)));
typedef float v8f __attribute__((ext_vector_type(8)));

__device__ __forceinline__ v8f wmma_f32_4(v2f a, v2f b, v8f c) {
  // D = A(16x4,f32) * B(4x16,f32) + C(16x16,f32)
  return __builtin_amdgcn_wmma_f32_16x16x4_f32(false, a, false, b,
                                               (short)0, c, false, false);
}

// ---------------------------------------------------------------------------
// Kernel 1: QKV projection.  Y(T x 384) = X(T x 128) @ Wqkv^T, scattered into
// q/k/v with layouts [b][h][s][d] (sequence) and [h][p][d] (patch).
// One wave per 16x16 output tile; 2050 M-tiles x 24 N-tiles.
// ---------------------------------------------------------------------------
__global__ __launch_bounds__(256) void qkv_kernel(
    const float* __restrict__ xp, const float* __restrict__ xs,
    const float* __restrict__ Wqkv, float* __restrict__ ws)
{
  const int lane = threadIdx.x & 31;
  const int warp = threadIdx.x >> 5;
  const int l16  = lane & 15;
  const int hi   = lane >> 4;

  const int tile = blockIdx.x * 8 + warp;       // 6150*8 == 49200 exact
  const int mt = tile / 24, nt = tile % 24;
  const int m0 = mt * 16, n0 = nt * 16;

  const float* arow = (m0 >= TS) ? (xp + (m0 - TS + l16) * DIM)
                                 : (xs + (m0 + l16) * DIM);
  const float* brow = Wqkv + (n0 + l16) * DIM;  // B[k][n] = Wqkv[n][k]

  v8f c = {};
  #pragma unroll
  for (int k0 = 0; k0 < DIM; k0 += 4) {
    v2f a = *(const v2f*)(arow + k0 + 2 * hi);
    v2f b = *(const v2f*)(brow + k0 + 2 * hi);
    c = wmma_f32_4(a, b, c);
  }

  #pragma unroll
  for (int j = 0; j < 8; ++j) {
    const int r   = m0 + j + 8 * hi;            // virtual token row
    const int f   = n0 + l16;                   // qkv feature 0..383
    const int i   = f >> 7;                     // 0=q, 1=k, 2=v
    const int rem = f & 127;
    const int h   = rem >> 4;
    const int d   = rem & 15;
    const float val = c[j];
    if (r < TS) {
      const int s = r >> 6, b = r & 63;         // r = s*B + b
      const int base = (i == 0) ? QS_OFF : (i == 1) ? KS_OFF : VS_OFF;
      ws[base + ((b * H + h) * S + s) * DH + d] = val;
    } else {
      const int p = r - TS;
      const int base = (i == 0) ? QP_OFF : (i == 1) ? KP_OFF : VP_OFF;
      ws[base + (h * P + p) * DH + d] = val;
    }
  }
}

// ---------------------------------------------------------------------------
// Kernel 2: fused flash-attention per (b,h) over F=544 tokens (rows 0..31 are
// the batch-shared patch tokens).  One wave per (b,h, 16-query-row tile).
// Online softmax; scores never leave registers.  LDS transposes the P tile
// from C/D layout into A layout for the P x V WMMA chain.
// ---------------------------------------------------------------------------
__global__ __launch_bounds__(256) void attn_kernel(float* __restrict__ ws)
{
  __shared__ float xpose[8 * 256];              // 16x16 f32 tile per wave

  const int lane = threadIdx.x & 31;
  const int warp = threadIdx.x >> 5;
  const int l16  = lane & 15;
  const int hi   = lane >> 4;

  const int wg   = blockIdx.x * 8 + warp;       // 2176*8 == 512*34 exact
  const int qt   = wg % 34;                     // query tile 0..33
  const int pair = wg / 34;
  const int h    = pair & 7;
  const int b    = pair >> 3;

  const float* qs = ws + QS_OFF + (b * H + h) * S * DH;
  const float* ks = ws + KS_OFF + (b * H + h) * S * DH;
  const float* vs = ws + VS_OFF + (b * H + h) * S * DH;
  const float* qp = ws + QP_OFF + h * P * DH;
  const float* kp = ws + KP_OFF + h * P * DH;
  const float* vp = ws + VP_OFF + h * P * DH;

  // Q tile (16 rows x 16 d) as four chained A fragments, loaded once.
  const float* qbase = (qt < 2) ? (qp + qt * 16 * DH) : (qs + (qt * 16 - P) * DH);
  const float* qrow  = qbase + l16 * DH;
  v2f aq[4];
  #pragma unroll
  for (int c = 0; c < 4; ++c) aq[c] = *(const v2f*)(qrow + c * 4 + 2 * hi);

  float m_i[8], l_i[8];
  v8f acc = {};
  #pragma unroll
  for (int j = 0; j < 8; ++j) { m_i[j] = -1e30f; l_i[j] = 0.0f; }

  float* xw = xpose + warp * 256;
  constexpr float LOG2E = 1.4426950408889634f;

  for (int kt = 0; kt < 34; ++kt) {
    const float* kbase = (kt < 2) ? (kp + kt * 16 * DH) : (ks + (kt * 16 - P) * DH);
    const float* vbase = (kt < 2) ? (vp + kt * 16 * DH) : (vs + (kt * 16 - P) * DH);
    if (kt + 1 < 34) {  // hint next K tile toward the caches (global_prefetch_b8)
      const float* knxt = (kt + 1 < 2) ? (kp + (kt + 1) * 16 * DH)
                                       : (ks + ((kt + 1) * 16 - P) * DH);
      __builtin_prefetch(knxt + lane * 8, 0, 1);
    }

    // S = Q * K^T : B[d][n] = K[token n][d]
    const float* krow = kbase + l16 * DH;
    v8f sc = {};
    #pragma unroll
    for (int c = 0; c < 4; ++c) {
      v2f bk = *(const v2f*)(krow + c * 4 + 2 * hi);
      sc = wmma_f32_4(aq[c], bk, sc);
    }

    // Online softmax (rows live across the 16-lane half; row j / 8+j per VGPR)
    #pragma unroll
    for (int j = 0; j < 8; ++j) {
      float sv = sc[j] * 0.25f;                 // 1/sqrt(DH)
      float rm = sv;
      rm = fmaxf(rm, __shfl_xor(rm, 1, 32));
      rm = fmaxf(rm, __shfl_xor(rm, 2, 32));
      rm = fmaxf(rm, __shfl_xor(rm, 4, 32));
      rm = fmaxf(rm, __shfl_xor(rm, 8, 32));
      const float mn   = fmaxf(m_i[j], rm);
      const float corr = exp2f((m_i[j] - mn) * LOG2E);
      const float pv   = exp2f((sv - mn) * LOG2E);
      float rs = pv;
      rs += __shfl_xor(rs, 1, 32);
      rs += __shfl_xor(rs, 2, 32);
      rs += __shfl_xor(rs, 4, 32);
      rs += __shfl_xor(rs, 8, 32);
      l_i[j] = l_i[j] * corr + rs;
      acc[j] *= corr;
      m_i[j] = mn;
      xw[(j + 8 * hi) * 16 + l16] = pv;         // stage for transpose
    }

    // acc += P * V   (P re-read from LDS in A layout; intra-wave LDS is in-order)
    #pragma unroll
    for (int c = 0; c < 4; ++c) {
      v2f ap = *(const v2f*)(xw + l16 * 16 + c * 4 + 2 * hi);
      v2f bv;
      bv.x = vbase[(c * 4 + 2 * hi) * DH + l16];
      bv.y = vbase[(c * 4 + 2 * hi + 1) * DH + l16];
      acc = wmma_f32_4(ap, bv, acc);
    }
  }

  // Normalize and write pre-projection output: rows f*B + b, cols h*16 + d.
  #pragma unroll
  for (int j = 0; j < 8; ++j) {
    const int f = qt * 16 + j + 8 * hi;
    const int e = h * DH + l16;
    ws[ATT_OFF + (f * Bn + b) * DIM + e] = acc[j] / l_i[j];
  }
}

// ---------------------------------------------------------------------------
// Kernel 3: output projection  (F*B x 128) @ W0^T + bias -> d_out (F,B,DIM).
// ---------------------------------------------------------------------------
__global__ __launch_bounds__(256) void proj_kernel(
    const float* __restrict__ ws, const float* __restrict__ W0w,
    const float* __restrict__ W0b, float* __restrict__ out)
{
  const int lane = threadIdx.x & 31;
  const int warp = threadIdx.x >> 5;
  const int l16  = lane & 15;
  const int hi   = lane >> 4;

  const int wg = blockIdx.x * 8 + warp;         // 2176*8 == 2176*8 exact
  const int mt = wg >> 3, nt = wg & 7;
  const int m0 = mt * 16, n0 = nt * 16;

  const float* arow = ws + ATT_OFF + (m0 + l16) * DIM;
  const float* brow = W0w + (n0 + l16) * DIM;   // B[k][n] = W0w[n][k]

  v8f c = {};
  #pragma unroll
  for (int k0 = 0; k0 < DIM; k0 += 4) {
    v2f a = *(const v2f*)(arow + k0 + 2 * hi);
    v2f b = *(const v2f*)(brow + k0 + 2 * hi);
    c = wmma_f32_4(a, b, c);
  }

  const float bias = W0b[n0 + l16];
  #pragma unroll
  for (int j = 0; j < 8; ++j) {
    const int r = m0 + j + 8 * hi;
    out[r * DIM + n0 + l16] = c[j] + bias;
  }
}

// ---------------------------------------------------------------------------
extern "C" void kernel_launch(void* const* d_in, const int* in_sizes, int n_in,
                              void* d_out, int out_size, void* d_ws, size_t ws_size,
                              hipStream_t stream) {
  const float* xp   = (const float*)d_in[0];   // (32, 1, 128)
  const float* xs   = (const float*)d_in[1];   // (512, 64, 128)
  const float* Wqkv = (const float*)d_in[2];   // (384, 128)
  const float* W0w  = (const float*)d_in[3];   // (128, 128)
  const float* W0b  = (const float*)d_in[4];   // (128,)
  float* ws  = (float*)d_ws;
  float* out = (float*)d_out;

  // 2050 M-tiles * 24 N-tiles = 49200 waves / 8 waves per block
  qkv_kernel<<<6150, 256, 0, stream>>>(xp, xs, Wqkv, ws);
  // 512 (b,h) pairs * 34 query tiles = 17408 waves / 8
  attn_kernel<<<2176, 256, 0, stream>>>(ws);
  // 2176 M-tiles * 8 N-tiles = 17408 waves / 8
  proj_kernel<<<2176, 256, 0, stream>>>(ws, W0w, W0b, out);
}